// Matching_5634997092735
// MI455X (gfx1250) — compile-verified
//
#include <hip/hip_runtime.h>

typedef __attribute__((ext_vector_type(2))) float v2f;
typedef __attribute__((ext_vector_type(8))) float v8f;

#define C_IN    64
#define C_OUT   16
#define D_MAX   48
#define ND      (D_MAX + 1)      // 49
#define HROWS   256
#define WCOLS   512
#define RPW     (WCOLS + D_MAX)  // 560 (48-col zero pad on the left)
#define NTHREADS 1024

__global__ __launch_bounds__(NTHREADS)
void cost_volume_wmma_kernel(const float* __restrict__ left,
                             const float* __restrict__ right,
                             const float* __restrict__ Wop,   // (16, 128) row-major
                             const float* __restrict__ bop,   // (16,)
                             float* __restrict__ out)         // (16, 49, 256, 512)
{
    // LDS staging: transposed weights [c][o], projected rows, bias
    __shared__ __align__(16) float lds_Wt[2 * C_IN * C_OUT];  // 2048 f  (8 KB)
    __shared__ __align__(16) float lds_L [C_OUT * WCOLS];     // 8192 f (32 KB)
    __shared__ __align__(16) float lds_R [C_OUT * RPW];       // 8960 f (35 KB)
    __shared__ float lds_b[C_OUT];

    const int tid = threadIdx.x;
    const int y   = blockIdx.x;

    // ---- Stage weights (transposed: lds_Wt[c*16+o] = W[o][c]), bias, zero R pad
    for (int i = tid; i < 2 * C_IN * C_OUT; i += NTHREADS) {
        const int o = i >> 7;          // /128
        const int c = i & 127;
        lds_Wt[c * C_OUT + o] = Wop[i];
    }
    if (tid < C_OUT) lds_b[tid] = bop[tid];
    for (int i = tid; i < C_OUT * D_MAX; i += NTHREADS) {      // 768 zeros
        const int o = i / D_MAX;
        const int p = i - o * D_MAX;
        lds_R[o * RPW + p] = 0.0f;
    }
    __syncthreads();

    // ---- Phase 1: L = W_l * left_row, R = W_r * right_row via V_WMMA_F32_16X16X4_F32
    // 64 jobs: j in [0,32) -> L tile (x0 = 16*j), j in [32,64) -> R tile.
    const int wave = tid >> 5;            // 0..31
    const int lane = tid & 31;
    const int n    = lane & 15;           // A: row M (=o) / B: col N (=x) / D: col N
    const int koff = (lane >> 4) << 1;    // lanes 0-15 hold K{0,1}; 16-31 hold K{2,3}
    const int obase = (lane >> 4) * 8;    // D rows: v -> M=v (lo lanes) / v+8 (hi lanes)

    for (int j = wave; j < 64; j += 32) {
        const bool   isR   = (j >= 32);
        const float* emb   = isR ? right : left;
        const int    cbase = isR ? C_IN : 0;
        const int    x0    = (j & 31) * 16;
        v8f acc = {};
        #pragma unroll
        for (int k = 0; k < C_IN / 4; ++k) {          // 16 WMMA steps, K=4 each
            const int c0 = 4 * k + koff;
            v2f a, b;
            a.x = lds_Wt[(cbase + c0    ) * C_OUT + n];
            a.y = lds_Wt[(cbase + c0 + 1) * C_OUT + n];
            const float* p = emb + (size_t)c0 * (HROWS * WCOLS)
                                 + (size_t)y * WCOLS + x0 + n;
            b.x = p[0];
            b.y = p[HROWS * WCOLS];
            acc = __builtin_amdgcn_wmma_f32_16x16x4_f32(
                      false, a, false, b, (short)0, acc, false, false);
        }
        if (!isR) {
            #pragma unroll
            for (int v = 0; v < 8; ++v)
                lds_L[(obase + v) * WCOLS + x0 + n] = acc[v];
        } else {
            #pragma unroll
            for (int v = 0; v < 8; ++v)
                lds_R[(obase + v) * RPW + D_MAX + x0 + n] = acc[v];
        }
    }
    __syncthreads();

    // ---- Phase 2: out[o,d,y,x] = L[o,x] + Rpad[o, 48-d+x] + b[o], float4 stores
    for (int o = 0; o < C_OUT; ++o) {
        const float  bo   = lds_b[o];
        const float* Lrow = &lds_L[o * WCOLS];
        const float* Rrow = &lds_R[o * RPW];
        float* orow = out + (((size_t)o * ND) * HROWS + y) * WCOLS;
        for (int idx = tid; idx < ND * (WCOLS / 4); idx += NTHREADS) {  // 6272 items
            const int d = idx >> 7;        // /128
            const int q = idx & 127;
            const int x = q * 4;
            const float4 lf = *reinterpret_cast<const float4*>(Lrow + x);
            const int rb = D_MAX - d + x;  // >= 0 always; shift breaks 16B align -> scalars
            float4 res;
            res.x = lf.x + Rrow[rb + 0] + bo;
            res.y = lf.y + Rrow[rb + 1] + bo;
            res.z = lf.z + Rrow[rb + 2] + bo;
            res.w = lf.w + Rrow[rb + 3] + bo;
            *reinterpret_cast<float4*>(orow + (size_t)d * (HROWS * WCOLS) + x) = res;
        }
    }
}

extern "C" void kernel_launch(void* const* d_in, const int* in_sizes, int n_in,
                              void* d_out, int out_size, void* d_ws, size_t ws_size,
                              hipStream_t stream) {
    (void)in_sizes; (void)n_in; (void)d_ws; (void)ws_size; (void)out_size;
    const float* left  = (const float*)d_in[0];
    const float* right = (const float*)d_in[1];
    const float* Wop   = (const float*)d_in[2];
    const float* bop   = (const float*)d_in[3];
    float* out = (float*)d_out;

    cost_volume_wmma_kernel<<<HROWS, NTHREADS, 0, stream>>>(left, right, Wop, bop, out);
}